// MultiHeadAttention_17849884082432
// MI455X (gfx1250) — compile-verified
//
#include <hip/hip_runtime.h>
#include <hip/hip_bf16.h>

#define EMBED   512
#define NHEADS  8
#define HDIM    64
#define SEQ     1024
#define BSZ     8
#define SCALE   0.125f   /* 64^-0.5 */

typedef float v2f __attribute__((ext_vector_type(2)));
typedef float v4f __attribute__((ext_vector_type(4)));
typedef float v8f __attribute__((ext_vector_type(8)));

// D = A(16x4) * B(4x16) + C, full f32 (keeps reference fp32 numerics).
__device__ __forceinline__ v8f wmma_f32_k4(v2f a, v2f b, v8f c) {
  return __builtin_amdgcn_wmma_f32_16x16x4_f32(
      /*neg_a=*/false, a, /*neg_b=*/false, b,
      /*c_mod=*/(short)0, c, /*reuse_a=*/false, /*reuse_b=*/false);
}

// ---- CDNA5 async global->LDS staging (ASYNCcnt path, no VGPR bounce) ------
// Per CDNA5 ISA: each lane supplies its own LDS byte address in
// VDST and its own global address in VADDR. Low 32 bits of a generic pointer
// to LDS are the LDS offset (aperture occupies bits [63:32]).
__device__ __forceinline__ void async_b128(void* lds, const void* g) {
  unsigned off = (unsigned)(unsigned long long)lds;
  asm volatile("global_load_async_to_lds_b128 %0, %1, off"
               :: "v"(off), "v"(g) : "memory");
}
__device__ __forceinline__ void async_b32(void* lds, const void* g) {
  unsigned off = (unsigned)(unsigned long long)lds;
  asm volatile("global_load_async_to_lds_b32 %0, %1, off"
               :: "v"(off), "v"(g) : "memory");
}
__device__ __forceinline__ void wait_async0() {
  asm volatile("s_wait_asynccnt 0" ::: "memory");
}

// ---------------------------------------------------------------------------
// Stage helpers: tiles are [rows][36] padded f32 in LDS.
// ---------------------------------------------------------------------------
__device__ __forceinline__ void stage_A128(float (*As)[36],
                                           const float* __restrict__ Ablk,
                                           int lda, int k0) {
  const int t = threadIdx.x;
  #pragma unroll
  for (int i = 0; i < 4; ++i) {
    int f = t + 256 * i;            // 0..1023 float4 chunks of 128x32 tile
    int r = f >> 3;
    int c = (f & 7) << 2;
    async_b128(&As[r][c], Ablk + (size_t)r * lda + k0 + c);
  }
}
__device__ __forceinline__ void stage_B64_rowmajor(float (*Bs)[36],
                                                   const float* __restrict__ Bblk,
                                                   int ldbn, int k0) {
  const int t = threadIdx.x;
  #pragma unroll
  for (int i = 0; i < 2; ++i) {
    int f = t + 256 * i;            // 0..511 float4 chunks of 64x32 tile
    int r = f >> 3;
    int c = (f & 7) << 2;
    async_b128(&Bs[r][c], Bblk + (size_t)r * ldbn + k0 + c);
  }
}
__device__ __forceinline__ void stage_B64_kmajor(float (*Bs)[36],
                                                 const float* __restrict__ Bblk,
                                                 int ldbk, int k0) {
  const int t = threadIdx.x;
  const int n = t & 63, ks = t >> 6;  // 64 lanes across n, 4 k-slices
  #pragma unroll
  for (int i = 0; i < 8; ++i) {
    int k = ks * 8 + i;
    async_b32(&Bs[n][k], Bblk + (size_t)(k0 + k) * ldbk + n);
  }
}

// ---------------------------------------------------------------------------
// 128x64 tile GEMM core, double-buffered LDS, async staging.
// 256 threads = 8 waves; wave w owns rows [w*16, w*16+16) x all 64 cols
// (4 accumulators). C = A(128xK) * B^T with B rows over n (ldbk==1) or
// k-major B (attn*V case).
// ---------------------------------------------------------------------------
__device__ __forceinline__ void gemm_tile_128x64(
    const float* __restrict__ Ablk, int lda,
    const float* __restrict__ Bblk, int ldbn, int ldbk, int Kdim,
    float (*As)[128][36], float (*Bs)[64][36], v8f acc[4])
{
  const int lane = threadIdx.x & 31;
  const int wave = threadIdx.x >> 5;
  const int fr   = lane & 15;
  const int kh   = (lane >> 4) << 1;

  stage_A128(As[0], Ablk, lda, 0);
  if (ldbk == 1) stage_B64_rowmajor(Bs[0], Bblk, ldbn, 0);
  else           stage_B64_kmajor (Bs[0], Bblk, ldbk, 0);
  wait_async0();
  __syncthreads();

  int cur = 0;
  for (int k0 = 0; k0 < Kdim; k0 += 32) {
    const int nxt = cur ^ 1;
    if (k0 + 32 < Kdim) {
      stage_A128(As[nxt], Ablk, lda, k0 + 32);
      if (ldbk == 1) stage_B64_rowmajor(Bs[nxt], Bblk, ldbn, k0 + 32);
      else           stage_B64_kmajor (Bs[nxt], Bblk, ldbk, k0 + 32);
    }
    #pragma unroll
    for (int k = 0; k < 32; k += 4) {
      v2f a = *(const v2f*)&As[cur][wave * 16 + fr][k + kh];
      #pragma unroll
      for (int tb = 0; tb < 4; ++tb) {
        v2f b = *(const v2f*)&Bs[cur][tb * 16 + fr][k + kh];
        acc[tb] = wmma_f32_k4(a, b, acc[tb]);
      }
    }
    wait_async0();     // my async writes into nxt are complete
    __syncthreads();   // everyone done reading cur + staging nxt
    cur = nxt;
  }
}

// ---------------------------------------------------------------------------
// K1: Q/K/V projections. Y = X * W^T. grid.z selects (Wq,Q),(Wk,K),(Wv,V).
// ---------------------------------------------------------------------------
__global__ void k_qkv(const float* __restrict__ x,
                      const float* __restrict__ Wq,
                      const float* __restrict__ Wk,
                      const float* __restrict__ Wv,
                      float* __restrict__ Q,
                      float* __restrict__ K,
                      float* __restrict__ V)
{
  __shared__ float As[2][128][36];
  __shared__ float Bs[2][64][36];
  const float* W; float* Y;
  if (blockIdx.z == 0)      { W = Wq; Y = Q; }
  else if (blockIdx.z == 1) { W = Wk; Y = K; }
  else                      { W = Wv; Y = V; }
  const int row0 = blockIdx.y * 128;
  const int col0 = blockIdx.x * 64;
  v8f acc[4] = {{}, {}, {}, {}};
  gemm_tile_128x64(x + (size_t)row0 * EMBED, EMBED,
                   W + (size_t)col0 * EMBED, EMBED, 1,
                   EMBED, As, Bs, acc);
  const int lane = threadIdx.x & 31, wave = threadIdx.x >> 5;
  const int cm = row0 + wave * 16 + ((lane >> 4) << 3);
  const int cn = col0 + (lane & 15);
  #pragma unroll
  for (int tb = 0; tb < 4; ++tb)
    #pragma unroll
    for (int r = 0; r < 8; ++r)
      Y[(size_t)(cm + r) * EMBED + cn + tb * 16] = acc[tb][r];
}

// ---------------------------------------------------------------------------
// K2: raw scores: S[b,h,i,j] = scale * <Q_i, K_j> + sum_c u[b,c,i,j]*Wu[h,c]
//     + bu[h], masked (-inf) where umask[b,j,i,0]. Written into attn region.
// ---------------------------------------------------------------------------
__global__ void k_scores(const float* __restrict__ Q,
                         const float* __restrict__ Kp,
                         const float* __restrict__ u,
                         const float* __restrict__ Wu,
                         const float* __restrict__ bu,
                         const unsigned char* __restrict__ umask,
                         float* __restrict__ attn)
{
  __shared__ float As[2][128][36];
  __shared__ float Bs[2][64][36];
  const int bh = blockIdx.z, b = bh >> 3, h = bh & 7;
  const int row0 = blockIdx.y * 128;
  const int col0 = blockIdx.x * 64;
  const float* Ab = Q  + ((size_t)b * SEQ + row0) * EMBED + h * HDIM;
  const float* Bb = Kp + ((size_t)b * SEQ + col0) * EMBED + h * HDIM;
  v8f acc[4] = {{}, {}, {}, {}};
  gemm_tile_128x64(Ab, EMBED, Bb, EMBED, 1, HDIM, As, Bs, acc);

  float wuc[6];
  #pragma unroll
  for (int c = 0; c < 6; ++c) wuc[c] = Wu[h * 6 + c];
  const float buh = bu[h];

  const int lane = threadIdx.x & 31, wave = threadIdx.x >> 5;
  const int im = row0 + wave * 16 + ((lane >> 4) << 3);
  const int jn = col0 + (lane & 15);
  float* C = attn + (size_t)bh * SEQ * SEQ;
  #pragma unroll
  for (int tb = 0; tb < 4; ++tb) {
    #pragma unroll
    for (int r = 0; r < 8; ++r) {
      const int i = im + r;
      const int j = jn + tb * 16;
      float s  = acc[tb][r] * SCALE;
      float ub = buh;
      #pragma unroll
      for (int c = 0; c < 6; ++c)
        ub += u[((size_t)(b * 6 + c) * SEQ + i) * SEQ + j] * wuc[c];
      float val = s + ub;
      if (umask[((size_t)b * SEQ + j) * SEQ + i]) val = -__builtin_inff();
      C[(size_t)i * SEQ + j] = val;
    }
  }
}

// ---------------------------------------------------------------------------
// K3: in-place row softmax. One wave32 per 1024-wide row, fully in registers:
// 32 f32/lane, max/sum via __shfl_xor butterfly (wave32).
// ---------------------------------------------------------------------------
__global__ void k_softmax(float* __restrict__ attn)
{
  const int lane = threadIdx.x & 31;
  const int wave = threadIdx.x >> 5;
  const size_t row = (size_t)blockIdx.x * 8 + wave;
  float* p = attn + row * SEQ;

  v4f v[8];
  #pragma unroll
  for (int c = 0; c < 8; ++c)
    v[c] = *(const v4f*)(p + c * 128 + lane * 4);

  float m = -__builtin_inff();
  #pragma unroll
  for (int c = 0; c < 8; ++c)
    m = fmaxf(m, fmaxf(fmaxf(v[c][0], v[c][1]), fmaxf(v[c][2], v[c][3])));
  #pragma unroll
  for (int off = 16; off >= 1; off >>= 1)
    m = fmaxf(m, __shfl_xor(m, off, 32));

  float s = 0.0f;
  #pragma unroll
  for (int c = 0; c < 8; ++c) {
    #pragma unroll
    for (int q = 0; q < 4; ++q) {
      float e = __expf(v[c][q] - m);
      v[c][q] = e;
      s += e;
    }
  }
  #pragma unroll
  for (int off = 16; off >= 1; off >>= 1)
    s += __shfl_xor(s, off, 32);

  const float inv = 1.0f / s;
  #pragma unroll
  for (int c = 0; c < 8; ++c) {
    v[c][0] *= inv; v[c][1] *= inv; v[c][2] *= inv; v[c][3] *= inv;
    *(v4f*)(p + c * 128 + lane * 4) = v[c];
  }
}

// ---------------------------------------------------------------------------
// K4: ctx[b,i,h,:] = sum_j attn[b,h,i,j] * V[b,j,h,:]   (B staged k-major)
// ---------------------------------------------------------------------------
__global__ void k_ctx(const float* __restrict__ attn,
                      const float* __restrict__ V,
                      float* __restrict__ ctx)
{
  __shared__ float As[2][128][36];
  __shared__ float Bs[2][64][36];
  const int bh = blockIdx.z, b = bh >> 3, h = bh & 7;
  const int row0 = blockIdx.y * 128;
  const float* Ab = attn + (size_t)bh * SEQ * SEQ + (size_t)row0 * SEQ;
  const float* Bb = V + (size_t)b * SEQ * EMBED + h * HDIM; // B[n][k] = Bb[k*EMBED + n]
  v8f acc[4] = {{}, {}, {}, {}};
  gemm_tile_128x64(Ab, SEQ, Bb, 1, EMBED, SEQ, As, Bs, acc);

  const int lane = threadIdx.x & 31, wave = threadIdx.x >> 5;
  const int im = row0 + wave * 16 + ((lane >> 4) << 3);
  const int nn = lane & 15;
  #pragma unroll
  for (int tb = 0; tb < 4; ++tb)
    #pragma unroll
    for (int r = 0; r < 8; ++r)
      ctx[((size_t)b * SEQ + im + r) * EMBED + h * HDIM + nn + tb * 16] = acc[tb][r];
}

// ---------------------------------------------------------------------------
// K5: out = ctx * Wo^T + bo
// ---------------------------------------------------------------------------
__global__ void k_out(const float* __restrict__ ctx,
                      const float* __restrict__ Wo,
                      const float* __restrict__ bo,
                      float* __restrict__ out)
{
  __shared__ float As[2][128][36];
  __shared__ float Bs[2][64][36];
  const int row0 = blockIdx.y * 128;
  const int col0 = blockIdx.x * 64;
  v8f acc[4] = {{}, {}, {}, {}};
  gemm_tile_128x64(ctx + (size_t)row0 * EMBED, EMBED,
                   Wo + (size_t)col0 * EMBED, EMBED, 1,
                   EMBED, As, Bs, acc);
  const int lane = threadIdx.x & 31, wave = threadIdx.x >> 5;
  const int cm = row0 + wave * 16 + ((lane >> 4) << 3);
  const int cn = col0 + (lane & 15);
  #pragma unroll
  for (int tb = 0; tb < 4; ++tb) {
    const float bb = bo[cn + tb * 16];
    #pragma unroll
    for (int r = 0; r < 8; ++r)
      out[(size_t)(cm + r) * EMBED + cn + tb * 16] = acc[tb][r] + bb;
  }
}

// ---------------------------------------------------------------------------
extern "C" void kernel_launch(void* const* d_in, const int* in_sizes, int n_in,
                              void* d_out, int out_size, void* d_ws, size_t ws_size,
                              hipStream_t stream)
{
  (void)in_sizes; (void)n_in; (void)out_size; (void)ws_size;
  const float*         x     = (const float*)d_in[0];
  const float*         u     = (const float*)d_in[1];
  const unsigned char* umask = (const unsigned char*)d_in[2]; // jnp bool_ = 1 byte
  const float*         Wq    = (const float*)d_in[3];
  const float*         Wk    = (const float*)d_in[4];
  const float*         Wv    = (const float*)d_in[5];
  const float*         Wu    = (const float*)d_in[6];
  const float*         bu    = (const float*)d_in[7];
  const float*         Wo    = (const float*)d_in[8];
  const float*         bo    = (const float*)d_in[9];

  float* out  = (float*)d_out;                                  // [8,1024,512]
  float* attn = out + (size_t)BSZ * SEQ * EMBED;                // [8,8,1024,1024]

  float* Q   = (float*)d_ws;                                    // 16 MB
  float* Kp  = Q  + (size_t)BSZ * SEQ * EMBED;                  // 16 MB
  float* V   = Kp + (size_t)BSZ * SEQ * EMBED;                  // 16 MB
  float* ctx = Q;                                               // reuse (Q dead after K2)

  const dim3 blk(256);
  k_qkv    <<<dim3(EMBED / 64, (BSZ * SEQ) / 128, 3),       blk, 0, stream>>>(x, Wq, Wk, Wv, Q, Kp, V);
  k_scores <<<dim3(SEQ / 64, SEQ / 128, BSZ * NHEADS),      blk, 0, stream>>>(Q, Kp, u, Wu, bu, umask, attn);
  k_softmax<<<dim3((BSZ * NHEADS * SEQ) / 8),               blk, 0, stream>>>(attn);
  k_ctx    <<<dim3(1, SEQ / 128, BSZ * NHEADS),             blk, 0, stream>>>(attn, V, ctx);
  k_out    <<<dim3(EMBED / 64, (BSZ * SEQ) / 128, 1),       blk, 0, stream>>>(ctx, Wo, bo, out);
}